// get_model_4715874091130
// MI455X (gfx1250) — compile-verified
//
#include <hip/hip_runtime.h>
#include <cstdint>
#include <cstddef>

#define BLK 256
#define B_  8
#define N1_ 4096
#define S1_ 512
#define S2_ 128

typedef float v2f __attribute__((ext_vector_type(2)));
typedef float v8f __attribute__((ext_vector_type(8)));

// ---------------------------------------------------------------- utilities
__global__ void zerof_kernel(float* p, long n) {
  long i = (long)blockIdx.x * blockDim.x + threadIdx.x;
  long stride = (long)gridDim.x * blockDim.x;
  for (; i < n; i += stride) p[i] = 0.0f;
}

__global__ void transpose_xyz_kernel(const float* __restrict__ xyz,
                                     float* __restrict__ coords,
                                     float* __restrict__ norm, int N) {
  long i = (long)blockIdx.x * blockDim.x + threadIdx.x;
  long total = (long)B_ * N;
  if (i >= total) return;
  int b = (int)(i / N), n = (int)(i % N);
  const float* src = xyz + (long)b * 6 * N + n;
  float* c = coords + i * 3;
  float* m = norm + i * 3;
  c[0] = src[0];        c[1] = src[(long)N];  c[2] = src[2L * N];
  m[0] = src[3L * N];   m[1] = src[4L * N];   m[2] = src[5L * N];
}

// pad weight rows (cout, cin) -> (cout, cinP) with zero fill
__global__ void pack_w_kernel(const float* __restrict__ src, float* __restrict__ dst,
                              int cout, int cin, int cinP) {
  long i = (long)blockIdx.x * blockDim.x + threadIdx.x;
  long total = (long)cout * cinP;
  long stride = (long)gridDim.x * blockDim.x;
  for (; i < total; i += stride) {
    int o = (int)(i / cinP), c = (int)(i % cinP);
    dst[i] = (c < cin) ? src[(long)o * cin + c] : 0.0f;
  }
}

// ---------------------------------------------------------------- FPS
// one workgroup per batch; sequential npoint iterations, LDS argmax
__global__ void __launch_bounds__(BLK) fps_kernel(const float* __restrict__ pts, int N,
                                                  int npoint, float* __restrict__ newxyz) {
  __shared__ float sd[BLK];
  __shared__ int   si[BLK];
  __shared__ int   sfar;
  int b = blockIdx.x, tid = threadIdx.x;
  const float* base = pts + (long)b * N * 3;
  float dmin[16];
  int nj = N >> 8;  // N is a multiple of 256 (4096 or 512)
  for (int j = 0; j < nj; ++j) dmin[j] = 1e10f;
  int far = 0;
  for (int it = 0; it < npoint; ++it) {
    float cx = base[3 * far], cy = base[3 * far + 1], cz = base[3 * far + 2];
    if (tid == 0) {
      float* o = newxyz + ((long)b * npoint + it) * 3;
      o[0] = cx; o[1] = cy; o[2] = cz;
    }
    float bd = -1.0f; int bi = 0;
    for (int j = 0; j < nj; ++j) {
      int n = tid + (j << 8);
      float dx = base[3 * n] - cx, dy = base[3 * n + 1] - cy, dz = base[3 * n + 2] - cz;
      float dd = dx * dx + dy * dy + dz * dz;
      float nd = fminf(dmin[j], dd);
      dmin[j] = nd;
      if (nd > bd) { bd = nd; bi = n; }  // ascending n per thread: keeps first max
    }
    sd[tid] = bd; si[tid] = bi;
    __syncthreads();
    for (int off = BLK >> 1; off > 0; off >>= 1) {
      if (tid < off) {
        float od = sd[tid + off]; int oi = si[tid + off];
        if (od > sd[tid] || (od == sd[tid] && oi < si[tid])) { sd[tid] = od; si[tid] = oi; }
      }
      __syncthreads();
    }
    if (tid == 0) sfar = si[0];
    __syncthreads();
    far = sfar;
  }
}

// ---------------------------------------------------------------- ball query
// one wave32 per center; first-K-in-index-order via ballot prefix counts
__global__ void query_ball_kernel(const float* __restrict__ pts,
                                  const float* __restrict__ centers,
                                  int* __restrict__ gidx, int N, int S, int K, float r2) {
  int wid = (int)(blockIdx.x * (blockDim.x >> 5) + (threadIdx.x >> 5));
  int lane = threadIdx.x & 31;
  if (wid >= B_ * S) return;
  int b = wid / S;
  const float* base = pts + (long)b * N * 3;
  const float* c = centers + (long)wid * 3;
  float cx = c[0], cy = c[1], cz = c[2];
  int* out = gidx + (long)wid * K;
  int count = 0;
  int firstn = 0x7fffffff;
  for (int nb = 0; nb < N && count < K; nb += 32) {
    int n = nb + lane;
    float dx = base[3 * n] - cx, dy = base[3 * n + 1] - cy, dz = base[3 * n + 2] - cz;
    bool valid = (dx * dx + dy * dy + dz * dz) <= r2;
    unsigned mask = (unsigned)__ballot(valid);
    int slot = count + __popc(mask & ((1u << lane) - 1u));
    if (valid && slot < K) {
      out[slot] = n;
      if (slot == 0) firstn = n;
    }
    count += __popc(mask);
  }
  if (count < K) {
    if (count == 0) firstn = N - 1;
    for (int off = 16; off > 0; off >>= 1) {
      int o = __shfl_xor(firstn, off);
      firstn = (o < firstn) ? o : firstn;
    }
    for (int j = count + lane; j < K; j += 32) out[j] = firstn;
  }
}

// ---------------------------------------------------------------- fused MLP
struct GatherArgs {
  const float* feats; const float* xyz; const float* centers; const int* gidx;
  int Cfeat, S, K, Npts, mode;   // mode 0: sa_msg (feat||relxyz), mode 1: group_all (xyz||feat)
  float* out; int outStride; int chanOff;
};
struct MlpArgs {
  const float *w0, *b0, *g0, *e0;
  const float *w1, *b1, *g1, *e1;
  const float *w2, *b2, *g2, *e2;
  int cin0P, c0, c1, c2, st0, st1;
};

// one 16(M=rows)x16(N=cout) output tile per wave iteration, reduce over cin via
// V_WMMA_F32_16X16X4_F32. f32 A (16x4) and B (4x16 = W^T) share the lane layout
//   val[v] = SRC[lane%16][kbase + (lane<16 ? v : 2+v)]   -> one b64 load each.
__device__ __forceinline__ void mlp_layer_wmma(const float* __restrict__ src, int ss, int cin,
    const float* __restrict__ w, int ws,
    const float* __restrict__ bb, const float* __restrict__ gg, const float* __restrict__ ee,
    float* __restrict__ dst, int ds, int cout) {
  int tid  = threadIdx.x;
  int lane = tid & 31;
  int wave = tid >> 5;
  int nw   = BLK >> 5;
  int l16  = lane & 15;
  int koff = (lane < 16) ? 0 : 2;
  int T = cout >> 4;
  for (int t = wave; t < T; t += nw) {
    int nbase = t << 4;
    v8f acc = {0.f, 0.f, 0.f, 0.f, 0.f, 0.f, 0.f, 0.f};
    const float* srow = src + l16 * ss + koff;
    const float* wrow = w + (long)(nbase + l16) * ws + koff;
    for (int cc = 0; cc < cin; cc += 4) {
      v2f av = *(const v2f*)(srow + cc);
      v2f bv = *(const v2f*)(wrow + cc);
      acc = __builtin_amdgcn_wmma_f32_16x16x4_f32(false, av, false, bv,
                                                  (short)0, acc, false, false);
    }
    int n = nbase + l16;
    float bia = bb[n], gam = gg[n], bet = ee[n];
    int mb = (lane < 16) ? 0 : 8;
#pragma unroll
    for (int r = 0; r < 8; ++r) {
      float v = gam * (acc[r] + bia) + bet;       // relu(gamma*(x@W^T+b)+beta)
      dst[(mb + r) * ds + n] = v > 0.0f ? v : 0.0f;
    }
  }
}

__global__ void __launch_bounds__(BLK) fused_mlp_kernel(GatherArgs ga, MlpArgs ma) {
  extern __shared__ float lds[];
  float* buf0 = lds;
  float* buf1 = lds + 16 * ma.st0;
  int chunks = ga.K >> 4;
  int chunk = blockIdx.x % chunks;
  int bs = blockIdx.x / chunks;       // b*S + s
  int b = bs / ga.S;
  int tid = threadIdx.x;
  int cinP = ma.cin0P;
  // gather 16 grouped rows into buf0 (zero-padded to cinP)
  for (int t = tid; t < 16 * cinP; t += BLK) {
    int r = t / cinP, c = t - r * cinP;
    int kg = (chunk << 4) + r;
    float v = 0.0f;
    if (ga.mode == 0) {
      int g = ga.gidx[(long)bs * ga.K + kg];
      if (c < ga.Cfeat)
        v = ga.feats[((long)b * ga.Npts + g) * ga.Cfeat + c];
      else if (c < ga.Cfeat + 3)
        v = ga.xyz[((long)b * ga.Npts + g) * 3 + (c - ga.Cfeat)]
          - ga.centers[(long)bs * 3 + (c - ga.Cfeat)];
    } else {
      if (c < 3)
        v = ga.xyz[((long)b * ga.Npts + kg) * 3 + c];
      else if (c < 3 + ga.Cfeat)
        v = ga.feats[((long)b * ga.Npts + kg) * ga.Cfeat + (c - 3)];
    }
    buf0[r * ma.st0 + c] = v;
  }
  __syncthreads();
  mlp_layer_wmma(buf0, ma.st0, ma.cin0P, ma.w0, ma.cin0P, ma.b0, ma.g0, ma.e0, buf1, ma.st1, ma.c0);
  __syncthreads();
  mlp_layer_wmma(buf1, ma.st1, ma.c0,    ma.w1, ma.c0,    ma.b1, ma.g1, ma.e1, buf0, ma.st0, ma.c1);
  __syncthreads();
  mlp_layer_wmma(buf0, ma.st0, ma.c1,    ma.w2, ma.c1,    ma.b2, ma.g2, ma.e2, buf1, ma.st1, ma.c2);
  __syncthreads();
  // max over the 16 rows of this chunk, accumulate across chunks via atomic max
  float* orow = ga.out + (long)bs * ga.outStride + ga.chanOff;
  for (int c = tid; c < ma.c2; c += BLK) {
    float m = buf1[c];
    for (int r = 1; r < 16; ++r) m = fmaxf(m, buf1[r * ma.st1 + c]);
    atomicMax((int*)(orow + c), __float_as_int(m));   // post-ReLU >= 0: int order ok
  }
}

// ---------------------------------------------------------------- FC head
__global__ void __launch_bounds__(BLK) head_kernel(const float* __restrict__ l3,
    const float* __restrict__ w1, const float* __restrict__ b1,
    const float* __restrict__ g1, const float* __restrict__ e1,
    const float* __restrict__ w2, const float* __restrict__ b2,
    const float* __restrict__ g2, const float* __restrict__ e2,
    const float* __restrict__ w3, const float* __restrict__ b3,
    float* __restrict__ out) {
  __shared__ float x1[512];
  __shared__ float x2[256];
  __shared__ float red[BLK];
  int b = blockIdx.x, tid = threadIdx.x;
  const float* x = l3 + (long)b * 1024;
  for (int o = tid; o < 512; o += BLK) {
    const float* w = w1 + (long)o * 1024;
    float s = b1[o];
    for (int c = 0; c < 1024; ++c) s += x[c] * w[c];
    s = g1[o] * s + e1[o];
    x1[o] = s > 0.f ? s : 0.f;
  }
  __syncthreads();
  for (int o = tid; o < 256; o += BLK) {
    const float* w = w2 + (long)o * 512;
    float s = b2[o];
    for (int c = 0; c < 512; ++c) s += x1[c] * w[c];
    s = g2[o] * s + e2[o];
    x2[o] = s > 0.f ? s : 0.f;
  }
  __syncthreads();
  red[tid] = (tid < 256) ? x2[tid] * w3[tid] : 0.f;
  __syncthreads();
  for (int off = BLK >> 1; off > 0; off >>= 1) {
    if (tid < off) red[tid] += red[tid + off];
    __syncthreads();
  }
  if (tid == 0) {
    float logit = red[0] + b3[0];
    out[b] = logit - logit;   // log_softmax over a single class == 0
  }
}

// ---------------------------------------------------------------- host
extern "C" void kernel_launch(void* const* d_in, const int* in_sizes, int n_in,
                              void* d_out, int out_size, void* d_ws, size_t ws_size,
                              hipStream_t stream) {
  (void)in_sizes; (void)n_in; (void)ws_size;
  auto F = [&](int i) { return (const float*)d_in[i]; };
  // jax tree-leaf order (sorted dict keys): 0..9 = bn1_b,bn1_g,bn2_b,bn2_g,
  // fc1_b,fc1_w,fc2_b,fc2_w,fc3_b,fc3_w; 10..45 sa1; 46..81 sa2; 82..93 sa3; 94 xyz.
  const float* xyz_in = F(94);

  const int M1[3][3] = {{32, 32, 64}, {64, 64, 128}, {64, 96, 128}};
  const int M2[3][3] = {{64, 64, 128}, {128, 128, 256}, {128, 128, 256}};
  const int M3[3]    = {256, 512, 1024};
  const int   K1[3] = {16, 32, 128};  const float R1[3] = {0.1f, 0.2f, 0.4f};
  const int   K2[3] = {32, 64, 128};  const float R2[3] = {0.2f, 0.4f, 0.8f};
  const int off1[3] = {0, 64, 192};   // sa1 branch channel offsets (total 320)
  const int off2[3] = {0, 128, 384};  // sa2 branch channel offsets (total 640)

  char* wp = (char*)d_ws;
  auto alloc = [&](size_t bytes) -> void* {
    void* p = (void*)wp;
    wp += (bytes + 255) & ~((size_t)255);
    return p;
  };
  float* coords = (float*)alloc(sizeof(float) * B_ * N1_ * 3);
  float* norm   = (float*)alloc(sizeof(float) * B_ * N1_ * 3);
  float* nx1    = (float*)alloc(sizeof(float) * B_ * S1_ * 3);
  float* nx2    = (float*)alloc(sizeof(float) * B_ * S2_ * 3);
  float* l1pts  = (float*)alloc(sizeof(float) * B_ * S1_ * 320);
  float* l2pts  = (float*)alloc(sizeof(float) * B_ * S2_ * 640);
  int* gidx1[3]; for (int i = 0; i < 3; ++i) gidx1[i] = (int*)alloc(sizeof(int) * B_ * S1_ * K1[i]);
  int* gidx2[3]; for (int i = 0; i < 3; ++i) gidx2[i] = (int*)alloc(sizeof(int) * B_ * S2_ * K2[i]);
  float* pw1[3]; for (int i = 0; i < 3; ++i) pw1[i] = (float*)alloc(sizeof(float) * M1[i][0] * 8);
  float* pw2[3]; for (int i = 0; i < 3; ++i) pw2[i] = (float*)alloc(sizeof(float) * M2[i][0] * 324);
  float* pw3 = (float*)alloc(sizeof(float) * M3[0] * 644);

  float* outF = (float*)d_out;       // [8 logits][8 x 1024 l3]
  float* l3   = outF + 8;

  // zero-init atomic-max targets and outputs
  zerof_kernel<<<64,  BLK, 0, stream>>>(outF, (long)out_size);
  zerof_kernel<<<256, BLK, 0, stream>>>(l1pts, (long)B_ * S1_ * 320);
  zerof_kernel<<<256, BLK, 0, stream>>>(l2pts, (long)B_ * S2_ * 640);

  { long tot = (long)B_ * N1_;
    transpose_xyz_kernel<<<(unsigned)((tot + BLK - 1) / BLK), BLK, 0, stream>>>(xyz_in, coords, norm, N1_); }

  // pack layer-0 weights (cin 6->8, 323->324, 643->644)
  for (int i = 0; i < 3; ++i)
    pack_w_kernel<<<8,   BLK, 0, stream>>>(F(10 + i * 12 + 3), pw1[i], M1[i][0], 6, 8);
  for (int i = 0; i < 3; ++i)
    pack_w_kernel<<<64,  BLK, 0, stream>>>(F(46 + i * 12 + 3), pw2[i], M2[i][0], 323, 324);
  pack_w_kernel<<<256, BLK, 0, stream>>>(F(82 + 3), pw3, M3[0], 643, 644);

  // -------- level 1
  fps_kernel<<<B_, BLK, 0, stream>>>(coords, N1_, S1_, nx1);
  for (int i = 0; i < 3; ++i) {
    int blocks = (B_ * S1_ + 7) / 8;
    query_ball_kernel<<<blocks, BLK, 0, stream>>>(coords, nx1, gidx1[i], N1_, S1_, K1[i], R1[i] * R1[i]);
  }
  for (int i = 0; i < 3; ++i) {
    int base = 10 + i * 12;
    MlpArgs ma;
    ma.w0 = pw1[i];       ma.b0 = F(base + 0);  ma.e0 = F(base + 1);  ma.g0 = F(base + 2);
    ma.w1 = F(base + 7);  ma.b1 = F(base + 4);  ma.e1 = F(base + 5);  ma.g1 = F(base + 6);
    ma.w2 = F(base + 11); ma.b2 = F(base + 8);  ma.e2 = F(base + 9);  ma.g2 = F(base + 10);
    ma.cin0P = 8; ma.c0 = M1[i][0]; ma.c1 = M1[i][1]; ma.c2 = M1[i][2];
    ma.st0 = ma.cin0P > ma.c1 ? ma.cin0P : ma.c1;
    ma.st1 = ma.c0 > ma.c2 ? ma.c0 : ma.c2;
    GatherArgs ga;
    ga.feats = norm; ga.xyz = coords; ga.centers = nx1; ga.gidx = gidx1[i];
    ga.Cfeat = 3; ga.S = S1_; ga.K = K1[i]; ga.Npts = N1_; ga.mode = 0;
    ga.out = l1pts; ga.outStride = 320; ga.chanOff = off1[i];
    unsigned grid = (unsigned)(B_ * S1_ * (K1[i] / 16));
    size_t shmem = (size_t)16 * (ma.st0 + ma.st1) * sizeof(float);
    fused_mlp_kernel<<<grid, BLK, shmem, stream>>>(ga, ma);
  }

  // -------- level 2
  fps_kernel<<<B_, BLK, 0, stream>>>(nx1, S1_, S2_, nx2);
  for (int i = 0; i < 3; ++i) {
    int blocks = (B_ * S2_ + 7) / 8;
    query_ball_kernel<<<blocks, BLK, 0, stream>>>(nx1, nx2, gidx2[i], S1_, S2_, K2[i], R2[i] * R2[i]);
  }
  for (int i = 0; i < 3; ++i) {
    int base = 46 + i * 12;
    MlpArgs ma;
    ma.w0 = pw2[i];       ma.b0 = F(base + 0);  ma.e0 = F(base + 1);  ma.g0 = F(base + 2);
    ma.w1 = F(base + 7);  ma.b1 = F(base + 4);  ma.e1 = F(base + 5);  ma.g1 = F(base + 6);
    ma.w2 = F(base + 11); ma.b2 = F(base + 8);  ma.e2 = F(base + 9);  ma.g2 = F(base + 10);
    ma.cin0P = 324; ma.c0 = M2[i][0]; ma.c1 = M2[i][1]; ma.c2 = M2[i][2];
    ma.st0 = ma.cin0P > ma.c1 ? ma.cin0P : ma.c1;
    ma.st1 = ma.c0 > ma.c2 ? ma.c0 : ma.c2;
    GatherArgs ga;
    ga.feats = l1pts; ga.xyz = nx1; ga.centers = nx2; ga.gidx = gidx2[i];
    ga.Cfeat = 320; ga.S = S2_; ga.K = K2[i]; ga.Npts = S1_; ga.mode = 0;
    ga.out = l2pts; ga.outStride = 640; ga.chanOff = off2[i];
    unsigned grid = (unsigned)(B_ * S2_ * (K2[i] / 16));
    size_t shmem = (size_t)16 * (ma.st0 + ma.st1) * sizeof(float);
    fused_mlp_kernel<<<grid, BLK, shmem, stream>>>(ga, ma);
  }

  // -------- sa3 (group all) -> l3 region of d_out
  {
    int base = 82;
    MlpArgs ma;
    ma.w0 = pw3;          ma.b0 = F(base + 0);  ma.e0 = F(base + 1);  ma.g0 = F(base + 2);
    ma.w1 = F(base + 7);  ma.b1 = F(base + 4);  ma.e1 = F(base + 5);  ma.g1 = F(base + 6);
    ma.w2 = F(base + 11); ma.b2 = F(base + 8);  ma.e2 = F(base + 9);  ma.g2 = F(base + 10);
    ma.cin0P = 644; ma.c0 = 256; ma.c1 = 512; ma.c2 = 1024;
    ma.st0 = 644;   // max(644, 512)
    ma.st1 = 1024;  // max(256, 1024)
    GatherArgs ga;
    ga.feats = l2pts; ga.xyz = nx2; ga.centers = nullptr; ga.gidx = nullptr;
    ga.Cfeat = 640; ga.S = 1; ga.K = S2_; ga.Npts = S2_; ga.mode = 1;
    ga.out = l3; ga.outStride = 1024; ga.chanOff = 0;
    unsigned grid = (unsigned)(B_ * (S2_ / 16));
    size_t shmem = (size_t)16 * (ma.st0 + ma.st1) * sizeof(float);  // ~107 KB dynamic LDS
    fused_mlp_kernel<<<grid, BLK, shmem, stream>>>(ga, ma);
  }

  // -------- head
  head_kernel<<<B_, BLK, 0, stream>>>(l3, F(5), F(4), F(1), F(0),
                                      F(7), F(6), F(3), F(2), F(9), F(8), outF);
}